// Perturbation_57810259804667
// MI455X (gfx1250) — compile-verified
//
#include <hip/hip_runtime.h>

typedef __attribute__((ext_vector_type(2))) float v2f;
typedef __attribute__((ext_vector_type(8))) float v8f;

#define Bsz   128
#define Tlen  3000
#define Ilen  6
#define Nn    128
#define Olen  2
#define SIGMA_REC 0.15f

#define XPITCH 132   // 128 + 4 pad -> bank = (4m + k) % 64, conflict-free
#define UPITCH 10    // 8 + 2 pad, keeps float2 (b64) alignment per row

// LDS-only workgroup barrier: fence LDS (dscnt) but NOT global stores/loads,
// so the per-step states stores / noise loads never serialize the scan.
#define WG_BARRIER() asm volatile(      \
    "s_wait_dscnt 0x0\n\t"              \
    "s_barrier_signal -1\n\t"           \
    "s_barrier_wait -1" ::: "memory")

static __device__ __forceinline__ v8f wmma_f32(v2f a, v2f b, v8f c) {
    return __builtin_amdgcn_wmma_f32_16x16x4_f32(false, a, false, b,
                                                 (short)0, c, false, false);
}

// ---------------------------------------------------------------------------
// Kernel 1: sequential scan. grid = 8 blocks (one 16-row batch tile each),
// 256 threads = 8 waves; wave w owns output columns [16w, 16w+16).
// W_rec^T slice resident in 64 VGPRs per wave; double-buffered LDS state with
// static addresses (loop unrolled by 2), one LDS-scope barrier per step,
// noise/u prefetched one step ahead through pointer accumulators.
// ---------------------------------------------------------------------------
__global__ __launch_bounds__(256) void rnn_scan_kernel(
    const float* __restrict__ u,      // [B,T,I]
    const float* __restrict__ p,      // [B,N]
    const float* __restrict__ alpha,  // [1]
    const float* __restrict__ noise,  // [B,T,N]
    const float* __restrict__ W_rec,  // [N,N]
    const float* __restrict__ W_in,   // [N,I]
    float* __restrict__ states)       // [B,T,N]
{
    __shared__ float xbufA[16 * XPITCH];
    __shared__ float xbufB[16 * XPITCH];
    __shared__ float suA[16 * UPITCH];
    __shared__ float suB[16 * UPITCH];

    const int tid     = threadIdx.x;
    const int wave    = tid >> 5;
    const int lane    = tid & 31;
    const int l       = lane & 15;        // N (and A-matrix M) index in tile
    const int hi      = lane >> 4;        // lane half selector
    const int rowBase = blockIdx.x * 16;  // batch rows of this workgroup
    const int colBase = wave * 16;        // output columns of this wave

    const float a           = alpha[0];
    const float scale       = sqrtf(2.0f * a * SIGMA_REC * SIGMA_REC);
    const float one_minus_a = 1.0f - a;

    // ---- Resident B fragments: W_rec^T[k, colBase+l], K in blocks of 4 ----
    // B 4x16 layout: VGPR0 = K = k0+2*hi, VGPR1 = K = k0+1+2*hi, N = lane%16
    v2f bfrag[32];
    {
        const float* wrow = W_rec + (size_t)(colBase + l) * Nn;
        #pragma unroll
        for (int ks = 0; ks < 32; ++ks) {
            bfrag[ks].x = wrow[4 * ks + 2 * hi];
            bfrag[ks].y = wrow[4 * ks + 2 * hi + 1];
        }
    }
    // W_in^T padded from K=6 to K=8 -> 2 extra B fragments
    v2f binfrag[2];
    {
        const float* wirow = W_in + (size_t)(colBase + l) * Ilen;
        #pragma unroll
        for (int ks = 0; ks < 2; ++ks) {
            int k = 4 * ks + 2 * hi;
            binfrag[ks].x = (k     < Ilen) ? wirow[k]     : 0.0f;
            binfrag[ks].y = (k + 1 < Ilen) ? wirow[k + 1] : 0.0f;
        }
    }

    // ---- p bias in C/D layout: VGPR r <-> (M = r + 8*hi, N = l) ----
    float pC[8];
    #pragma unroll
    for (int r = 0; r < 8; ++r)
        pC[r] = p[(size_t)(rowBase + r + 8 * hi) * Nn + colBase + l];

    // ---- per-lane global bases (r adds a compile-time Tlen*Nn multiple) ----
    const size_t rstride = (size_t)Tlen * Nn;               // per-row stride
    const float* nzbase  = noise  + (size_t)(rowBase + 8 * hi) * rstride + colBase + l;
    float*       stbase  = states + (size_t)(rowBase + 8 * hi) * rstride + colBase + l;

    // ---- u staging: 96 threads own one (m, k) cell each, pointer-bumped ----
    const bool stager = tid < 96;
    const int  sm = tid / 6;              // row within tile   (valid if stager)
    const int  sk = tid - sm * 6;         // input channel     (valid if stager)
    const float* up = u + ((size_t)(rowBase + (stager ? sm : 0)) * Tlen + 1) * Ilen
                        + (stager ? sk : 0);                // points at t+1 = 1

    // ---- init: x = 0, states[:,0,:] = 0, stage suA for t=0, preload noise ----
    float xC[8];
    #pragma unroll
    for (int r = 0; r < 8; ++r) xC[r] = 0.0f;
    #pragma unroll
    for (int r = 0; r < 8; ++r)
        xbufA[(r + 8 * hi) * XPITCH + colBase + l] = 0.0f;
    #pragma unroll
    for (int r = 0; r < 8; ++r)
        stbase[(size_t)r * rstride] = 0.0f;
    if (stager)
        suA[sm * UPITCH + sk] = u[((size_t)(rowBase + sm) * Tlen) * Ilen + sk];
    if (tid < 32) {                       // K-pad cells (k = 6,7) stay 0 forever
        int mm = tid & 15, kb = tid >> 4;
        suA[mm * UPITCH + 6 + kb] = 0.0f;
        suB[mm * UPITCH + 6 + kb] = 0.0f;
    }
    float nzr[8];                         // noise for the step being computed
    #pragma unroll
    for (int r = 0; r < 8; ++r)
        nzr[r] = nzbase[(size_t)r * rstride];

    size_t off = Nn;                      // element offset of timestep t+1

    WG_BARRIER();

    auto step = [&](const float* xb_r, const float* su_r,
                    float* xb_w, float* su_w) {
        // ---- prefetch next step's u and noise (independent of LDS) ----
        float uN = 0.0f;
        if (stager) { uN = *up; up += Ilen; }
        float nzN[8];
        #pragma unroll
        for (int r = 0; r < 8; ++r)
            nzN[r] = nzbase[(size_t)r * rstride + off];

        // ---- y = x @ W_rec^T + u @ W_in^T : 34 fp32 WMMAs, 4 acc chains ----
        v8f acc0 = {}, acc1 = {}, acc2 = {}, acc3 = {};
        const float* xrow = xb_r + l * XPITCH;
        #pragma unroll
        for (int g = 0; g < 8; ++g) {
            v2f a0 = *(const v2f*)(xrow + 4 * (4 * g + 0) + 2 * hi);
            v2f a1 = *(const v2f*)(xrow + 4 * (4 * g + 1) + 2 * hi);
            v2f a2 = *(const v2f*)(xrow + 4 * (4 * g + 2) + 2 * hi);
            v2f a3 = *(const v2f*)(xrow + 4 * (4 * g + 3) + 2 * hi);
            acc0 = wmma_f32(a0, bfrag[4 * g + 0], acc0);
            acc1 = wmma_f32(a1, bfrag[4 * g + 1], acc1);
            acc2 = wmma_f32(a2, bfrag[4 * g + 2], acc2);
            acc3 = wmma_f32(a3, bfrag[4 * g + 3], acc3);
        }
        {
            const float* urow = su_r + l * UPITCH;
            v2f a0 = *(const v2f*)(urow + 0 + 2 * hi);
            v2f a1 = *(const v2f*)(urow + 4 + 2 * hi);
            acc0 = wmma_f32(a0, binfrag[0], acc0);
            acc1 = wmma_f32(a1, binfrag[1], acc1);
        }

        // ---- x_new = (1-a)x + a*relu(y + p + scale*noise), C/D layout ----
        float xn[8];
        #pragma unroll
        for (int r = 0; r < 8; ++r) {
            float pre = (acc0[r] + acc1[r]) + (acc2[r] + acc3[r])
                        + pC[r] + scale * nzr[r];
            float rel = pre > 0.0f ? pre : 0.0f;
            xn[r]     = one_minus_a * xC[r] + a * rel;
        }

        // ---- write next-step buffers (other parity), stream states out ----
        #pragma unroll
        for (int r = 0; r < 8; ++r)
            xb_w[(r + 8 * hi) * XPITCH + colBase + l] = xn[r];
        if (stager) su_w[sm * UPITCH + sk] = uN;
        #pragma unroll
        for (int r = 0; r < 8; ++r) {
            stbase[(size_t)r * rstride + off] = xn[r];
            xC[r]  = xn[r];
            nzr[r] = nzN[r];
        }
        off += Nn;

        WG_BARRIER();   // LDS-only fence: writes visible, reads retired
    };

    // 2999 steps = 1499 statically-addressed A->B / B->A pairs + 1 epilogue
    for (int it = 0; it < (Tlen - 2) / 2; ++it) {
        step(xbufA, suA, xbufB, suB);
        step(xbufB, suB, xbufA, suA);
    }
    step(xbufA, suA, xbufB, suB);
}

// ---------------------------------------------------------------------------
// Kernel 2: out[b,t,:] = relu(states[b,t,:]) @ W_out^T.  One wave per (b,t),
// float4 coalesced reads, wave32 shuffle reduction.
// ---------------------------------------------------------------------------
__global__ __launch_bounds__(256) void rnn_out_kernel(
    const float* __restrict__ states,  // [B,T,N]
    const float* __restrict__ W_out,   // [O,N]
    float* __restrict__ out)           // [B,T,O]
{
    const int w    = blockIdx.x * 8 + (threadIdx.x >> 5);
    const int lane = threadIdx.x & 31;
    if (w >= Bsz * Tlen) return;

    const float4 s4 = *(const float4*)(states + (size_t)w * Nn + lane * 4);
    const float4 w0 = *(const float4*)(W_out + lane * 4);
    const float4 w1 = *(const float4*)(W_out + Nn + lane * 4);

    float rx = fmaxf(s4.x, 0.0f), ry = fmaxf(s4.y, 0.0f);
    float rz = fmaxf(s4.z, 0.0f), rw = fmaxf(s4.w, 0.0f);
    float p0 = rx * w0.x + ry * w0.y + rz * w0.z + rw * w0.w;
    float p1 = rx * w1.x + ry * w1.y + rz * w1.z + rw * w1.w;

    #pragma unroll
    for (int off = 16; off > 0; off >>= 1) {
        p0 += __shfl_xor(p0, off, 32);
        p1 += __shfl_xor(p1, off, 32);
    }
    if (lane == 0) {
        out[(size_t)w * Olen + 0] = p0;
        out[(size_t)w * Olen + 1] = p1;
    }
}

extern "C" void kernel_launch(void* const* d_in, const int* in_sizes, int n_in,
                              void* d_out, int out_size, void* d_ws, size_t ws_size,
                              hipStream_t stream) {
    const float* u     = (const float*)d_in[0];
    const float* p     = (const float*)d_in[1];
    const float* alpha = (const float*)d_in[2];
    const float* noise = (const float*)d_in[3];
    const float* W_rec = (const float*)d_in[4];
    const float* W_in  = (const float*)d_in[5];
    const float* W_out = (const float*)d_in[6];

    float* out    = (float*)d_out;                         // [B,T,O] first
    float* states = out + (size_t)Bsz * Tlen * Olen;       // then [B,T,N]

    rnn_scan_kernel<<<Bsz / 16, 256, 0, stream>>>(u, p, alpha, noise,
                                                  W_rec, W_in, states);

    const int nwaves = Bsz * Tlen;
    rnn_out_kernel<<<(nwaves + 7) / 8, 256, 0, stream>>>(states, W_out, out);
}